// HardContrastiveLossV2_16277926052335
// MI455X (gfx1250) — compile-verified
//
#include <hip/hip_runtime.h>
#include <hip/hip_bf16.h>
#include <math.h>
#include <stdint.h>

#define BETA_F 0.2f
#define INV_BETA 5.0f
#define EPS_F 1e-8f
#define B_N 8192
#define D_N 256
#define P_N 4096
#define BATCH_ROWS 512
#define NUM_BATCH ((2 * P_N) / BATCH_ROWS) // 16

typedef __attribute__((ext_vector_type(2))) float v2f;
typedef __attribute__((ext_vector_type(8))) float v8f;
typedef __attribute__((ext_vector_type(4))) int   v4i;

// Address-space pointer construction via integer round-trip:
//  - global flat address == AS(1) address
//  - LDS flat aperture keeps the LDS offset in the low 32 bits (ISA 10.2),
//    so truncating to u32 yields a valid AS(3) offset pointer.
#define AS_G_V4I(p) ((__attribute__((address_space(1))) v4i*)(uintptr_t)(p))
#define AS_L_V4I(p) ((__attribute__((address_space(3))) v4i*)(uint32_t)(uintptr_t)(p))

#if __has_builtin(__builtin_amdgcn_global_load_async_to_lds_b128) && \
    __has_builtin(__builtin_amdgcn_s_wait_asynccnt)
#define USE_ASYNC_LDS 1
#else
#define USE_ASYNC_LDS 0
#endif

// slot s in [0, 2P): s < P -> ii of pair s ; else jj of pair s-P
__device__ __forceinline__ int slot_row(const int* __restrict__ pairs, int s) {
    return (s < P_N) ? pairs[2 * s] : pairs[2 * (s - P_N) + 1];
}

// ---------------- 1) normalize rows: e = x / max(||x||, eps) ----------------
__global__ void k_normalize(const float* __restrict__ x, float* __restrict__ e) {
    __shared__ float red[D_N];
    const int row = blockIdx.x;
    const int t = threadIdx.x;
    float val = x[(size_t)row * D_N + t];
    red[t] = val * val;
    __syncthreads();
    for (int s = D_N / 2; s > 0; s >>= 1) {
        if (t < s) red[t] += red[t + s];
        __syncthreads();
    }
    float nrm = fmaxf(sqrtf(red[0]), EPS_F);
    e[(size_t)row * D_N + t] = val / nrm;
}

// ---------------- 2) pos[p] = exp(dot(e_ii, e_jj)/beta), one wave32/pair ----
__global__ void k_pos(const float* __restrict__ e, const int* __restrict__ pairs,
                      float* __restrict__ pos) {
    const int wave = threadIdx.x >> 5;
    const int lane = threadIdx.x & 31;
    const int p = blockIdx.x * 8 + wave;
    const int ii = pairs[2 * p];
    const int jj = pairs[2 * p + 1];
    const float* a = e + (size_t)ii * D_N;
    const float* b = e + (size_t)jj * D_N;
    float s = 0.f;
    for (int k = lane; k < D_N; k += 32) s += a[k] * b[k];
    for (int m = 16; m >= 1; m >>= 1) s += __shfl_xor(s, m, 32);
    if (lane == 0) pos[p] = expf(s * INV_BETA);
}

// ---------------- 3a) WMMA GEMM: 512 gathered rows x 8192 cols, exp'd -------
// block = 256 threads (8 waves); A-tile (16 rows x 256) staged in LDS once,
// via GLOBAL_LOAD_ASYNC_TO_LDS_B128 (ASYNCcnt) when available. Each wave
// computes two 16x16 tiles with V_WMMA_F32_16X16X4_F32. Grid (32, 32)/batch.
#define ASTR 260   // padded LDS row stride (floats): bank-conflict-free ds_load_b64
__global__ void __launch_bounds__(256)
k_gemm_exp(const float* __restrict__ e, const int* __restrict__ pairs,
           float* __restrict__ buf, int batchBase) {
    __shared__ float As[16 * ASTR];   // ~16.6 KB
    __shared__ int rowIdx[16];

    const int tid = threadIdx.x;
    if (tid < 16) rowIdx[tid] = slot_row(pairs, batchBase + blockIdx.x * 16 + tid);
    __syncthreads();

    // stage A: thread -> 16 consecutive floats (4 x b128)
    {
        const int m = tid >> 4;             // 0..15
        const int kb = (tid & 15) * 16;     // 0,16,...,240
        const float* src = e + (size_t)rowIdx[m] * D_N + kb;
        float* dst = &As[m * ASTR + kb];
#if USE_ASYNC_LDS
#pragma unroll
        for (int c = 0; c < 4; ++c)
            __builtin_amdgcn_global_load_async_to_lds_b128(
                AS_G_V4I(src + 4 * c), AS_L_V4I(dst + 4 * c), 0, 0);
        __builtin_amdgcn_s_wait_asynccnt(0);
#else
#pragma unroll
        for (int c = 0; c < 4; ++c)
            ((float4*)dst)[c] = ((const float4*)src)[c];
#endif
    }
    __syncthreads();

    const int lane = tid & 31;
    const int wave = tid >> 5;
    const int m16 = lane & 15;   // M (A) / N (B) within tile
    const int kh  = lane >> 4;   // K half-select

    const int n0 = blockIdx.y * 256 + wave * 32;  // first of this wave's 32 cols
    const int n  = n0 + m16;

    // A from LDS: As[m16][k0 + 2*kh + {0,1}]; B from global (B = e^T, same gather)
    const float* aLds = &As[m16 * ASTR + 2 * kh];
    const float* bp0 = e + (size_t)n * D_N + 2 * kh;
    const float* bp1 = e + (size_t)(n + 16) * D_N + 2 * kh;

    v8f c0 = {};
    v8f c1 = {};
#pragma unroll 4
    for (int k0 = 0; k0 < D_N; k0 += 4) {
        v2f a  = *(const v2f*)(aLds + k0);
        v2f b0 = *(const v2f*)(bp0 + k0);
        v2f b1 = *(const v2f*)(bp1 + k0);
        c0 = __builtin_amdgcn_wmma_f32_16x16x4_f32(false, a, false, b0,
                                                   (short)0, c0, false, false);
        c1 = __builtin_amdgcn_wmma_f32_16x16x4_f32(false, a, false, b1,
                                                   (short)0, c1, false, false);
    }

    // C/D layout: col N = lane%16, row M = 8*(lane/16) + vgpr
    const int rowLocalBase = blockIdx.x * 16 + 8 * kh;
#pragma unroll
    for (int v = 0; v < 8; ++v) {
        float* dst = buf + (size_t)(rowLocalBase + v) * B_N;
        dst[n0 + m16]      = expf(c0[v] * INV_BETA);
        dst[n0 + 16 + m16] = expf(c1[v] * INV_BETA);
    }
}

// ---------------- 3b) per-row: mask, bitonic sort in LDS, 0.8-quantile, S ---
__global__ void __launch_bounds__(1024)
k_row_select(const float* __restrict__ buf, const int* __restrict__ pairs,
             float* __restrict__ S, int batchBase) {
    __shared__ float v[B_N];     // 32 KB
    __shared__ float red[1024];
    const int t = threadIdx.x;
    const int rs = batchBase + blockIdx.x;
    const int r = slot_row(pairs, rs);

    const float* src = buf + (size_t)blockIdx.x * B_N;
    for (int i = t; i < B_N; i += 1024) v[i] = src[i];
    __syncthreads();

    // mask: diagonal + every pair position touching row r (all writes are 0)
    if (t == 0) v[r] = 0.0f;
    for (int p = t; p < P_N; p += 1024) {
        int ii = pairs[2 * p], jj = pairs[2 * p + 1];
        if (ii == r) v[jj] = 0.0f;
        if (jj == r) v[ii] = 0.0f;
    }
    __syncthreads();

    // bitonic sort ascending
    for (int k = 2; k <= B_N; k <<= 1) {
        for (int j = k >> 1; j > 0; j >>= 1) {
            for (int i = t; i < B_N; i += 1024) {
                int ixj = i ^ j;
                if (ixj > i) {
                    float a = v[i], b = v[ixj];
                    bool up = ((i & k) == 0);
                    if ((a > b) == up) { v[i] = b; v[ixj] = a; }
                }
            }
            __syncthreads();
        }
    }

    // quantile(0.8) at position 8191*0.8 = 6552.8 -> lerp, then sum >= thr
    float thr = v[6552] + 0.8f * (v[6553] - v[6552]);
    float s = 0.f;
    for (int i = t; i < B_N; i += 1024) {
        float x = v[i];
        if (x >= thr) s += x;
    }
    red[t] = s;
    __syncthreads();
    for (int st = 512; st > 0; st >>= 1) {
        if (t < st) red[t] += red[t + st];
        __syncthreads();
    }
    if (t == 0) S[rs] = red[0];
}

// ---------------- 4) pos threshold: sort 4096, act = (pos <= sorted[819]) ---
// pos_ij == pos_ji exactly (symmetric fp dot products), so kmin == count(sel)
// and act degenerates to sel.
__global__ void __launch_bounds__(1024)
k_pos_thr(const float* __restrict__ pos, int* __restrict__ act,
          float* __restrict__ acc) {
    __shared__ float v[P_N];     // 16 KB
    const int t = threadIdx.x;
    for (int i = t; i < P_N; i += 1024) v[i] = pos[i];
    __syncthreads();
    for (int k = 2; k <= P_N; k <<= 1) {
        for (int j = k >> 1; j > 0; j >>= 1) {
            for (int i = t; i < P_N; i += 1024) {
                int ixj = i ^ j;
                if (ixj > i) {
                    float a = v[i], b = v[ixj];
                    bool up = ((i & k) == 0);
                    if ((a > b) == up) { v[i] = b; v[ixj] = a; }
                }
            }
            __syncthreads();
        }
    }
    float thr = v[819]; // (4096-1)*0.2 = 819.0 exactly, no interpolation
    for (int i = t; i < P_N; i += 1024) act[i] = (pos[i] <= thr) ? 1 : 0;
    if (t == 0) *acc = 0.0f;
}

// ---------------- 5) loss = sum over active p of sum_q log1p(S/pos[p]) -----
__global__ void k_loss(const float* __restrict__ pos, const float* __restrict__ S,
                       const int* __restrict__ act, float* __restrict__ acc) {
    __shared__ float red[256];
    const int p = blockIdx.x;
    if (!act[p]) return; // uniform across block
    const int t = threadIdx.x;
    const float inv = 1.0f / pos[p];
    float s = 0.f;
    for (int q = t; q < P_N; q += 256)
        s += log1pf(S[q] * inv) + log1pf(S[P_N + q] * inv);
    red[t] = s;
    __syncthreads();
    for (int st = 128; st > 0; st >>= 1) {
        if (t < st) red[t] += red[t + st];
        __syncthreads();
    }
    if (t == 0) atomicAdd(acc, red[0]);
}

__global__ void k_finalize(const float* __restrict__ acc, float* __restrict__ out) {
    out[0] = acc[0] / (2.0f * (float)P_N);
}

// ---------------------------------------------------------------------------
extern "C" void kernel_launch(void* const* d_in, const int* in_sizes, int n_in,
                              void* d_out, int out_size, void* d_ws, size_t ws_size,
                              hipStream_t stream) {
    const float* x = (const float*)d_in[0];     // embeddings (B, D) f32
    const int* pairs = (const int*)d_in[1];     // positive_pairs (P, 2) i32
    float* out = (float*)d_out;

    // workspace layout (floats)
    float* e   = (float*)d_ws;                  // 8192*256   (8 MB)
    float* pos = e + (size_t)B_N * D_N;         // 4096
    float* S   = pos + P_N;                     // 8192
    int*   act = (int*)(S + 2 * P_N);           // 4096
    float* acc = (float*)(act + P_N);           // 1 (+pad)
    float* buf = acc + 64;                      // 512*8192   (16 MB)

    k_normalize<<<B_N, D_N, 0, stream>>>(x, e);
    k_pos<<<P_N / 8, 256, 0, stream>>>(e, pairs, pos);

    for (int b = 0; b < NUM_BATCH; ++b) {
        const int base = b * BATCH_ROWS;
        dim3 grid(BATCH_ROWS / 16, B_N / 256);  // (32, 32)
        k_gemm_exp<<<grid, 256, 0, stream>>>(e, pairs, buf, base);
        k_row_select<<<BATCH_ROWS, 1024, 0, stream>>>(buf, pairs, S, base);
    }

    k_pos_thr<<<1, 1024, 0, stream>>>(pos, act, acc);
    k_loss<<<P_N, 256, 0, stream>>>(pos, S, act, acc);
    k_finalize<<<1, 1, 0, stream>>>(acc, out);
}